// Fastsum_549755813987
// MI455X (gfx1250) — compile-verified
//
#include <hip/hip_runtime.h>

#define NPTS 4096
#define MPTS 4096
#define DIMD 32
#define PSL  16
#define KFTN 256
#define NSPLIT 8
#define CHUNK 64

typedef __attribute__((ext_vector_type(2))) float v2f;
typedef __attribute__((ext_vector_type(8))) float v8f;

__device__ __forceinline__ float wave_sum(float v) {
  #pragma unroll
  for (int o = 16; o > 0; o >>= 1) v += __shfl_xor(v, o, 32);
  return v;
}
__device__ __forceinline__ float wave_max(float v) {
  #pragma unroll
  for (int o = 16; o > 0; o >>= 1) v = fmaxf(v, __shfl_xor(v, o, 32));
  return v;
}

__device__ __constant__ float kTWO_PI  = 6.28318530717958647692f;
__device__ __constant__ float kTWO_PI2 = 19.739208802178716f;      // 2*pi^2
__device__ __constant__ float kLOGC    = -9.58359321025047f;       // 16*ln(pi) - lgamma(16)

// ---------------- K0: normalize slicing directions, init maxabs ----------------
__global__ void k_init(const float* __restrict__ xis, float* __restrict__ xisn,
                       unsigned* __restrict__ maxbits) {
  int tid  = threadIdx.x;           // 512 threads = 16 waves, one wave per slice p
  int p    = tid >> 5;
  int lane = tid & 31;
  float v = xis[p * DIMD + lane];
  float s = wave_sum(v * v);
  xisn[p * DIMD + lane] = v * rsqrtf(s);
  if (tid == 0) *maxbits = 0u;
}

// ---------------- K1: projections via f32 WMMA (16x16x4), raw (unscaled) -------
// One wave per 16-point tile: D[16p x 16n] = xis_norm[16 x 32] * pts_chunk^T[32 x 16]
__global__ void k_proj(const float* __restrict__ xisn, const float* __restrict__ x,
                       const float* __restrict__ y, float* __restrict__ xp,
                       float* __restrict__ yp, unsigned* __restrict__ maxbits) {
  const int lane = threadIdx.x;
  const int rc = lane & 15;         // A row (p) / B col (point) index
  const int hi = lane >> 4;         // half-wave selects K pair {0,1} vs {2,3}
  const bool isY = blockIdx.x >= (NPTS / 16);
  const int tile = isY ? (blockIdx.x - NPTS / 16) : blockIdx.x;
  const float* pts = isY ? y : x;
  float* dst = isY ? yp : xp;
  const int n0 = tile * 16;

  const float* arow = xisn + rc * DIMD;            // A[M=rc][k]
  const float* brow = pts + (size_t)(n0 + rc) * DIMD; // B[k][N=rc] = pts[n0+rc][k]

  v8f d = {};
  #pragma unroll
  for (int kk = 0; kk < 8; ++kk) {                 // 8 x K=4 covers K=32
    const int kb = kk * 4 + hi * 2;
    v2f a; a[0] = arow[kb]; a[1] = arow[kb + 1];
    v2f b; b[0] = brow[kb]; b[1] = brow[kb + 1];
    d = __builtin_amdgcn_wmma_f32_16x16x4_f32(false, a, false, b, (short)0, d,
                                              false, false);
  }

  float mx = 0.f;
  #pragma unroll
  for (int j = 0; j < 8; ++j) {                    // D: VGPR j -> M = j + 8*hi, N = rc
    const int prow = j + hi * 8;
    dst[(size_t)prow * NPTS + n0 + rc] = d[j];
    mx = fmaxf(mx, fabsf(d[j]));
  }
  mx = wave_max(mx);
  if (lane == 0) atomicMax(maxbits, __float_as_uint(mx));
}

// ---------------- K2: adjoint NUDFT with k = 16*k1 + k0 factorization ----------
// block (p, split); thread = k index. LDS tables E1[n][k0], E16[n][k1].
__global__ void k_stageA(const float* __restrict__ xp, const float* __restrict__ w,
                         float* __restrict__ apart,
                         const unsigned* __restrict__ maxbits) {
  __shared__ float2 sE1[CHUNK][16];
  __shared__ float2 sE16[CHUNK][16];
  __shared__ float  sW[CHUNK];

  const int p  = blockIdx.x;
  const int s  = blockIdx.y;
  const int kk = threadIdx.x;              // 0..255 -> k = kk-128 = 16*(k1i-8) + k0
  const int k0  = kk & 15;
  const int k1i = kk >> 4;
  const float sf = 0.3f / __uint_as_float(*maxbits);
  const int np = kk >> 2;                  // table-build: 64 points x 4 quarters
  const int q  = kk & 3;

  float accR = 0.f, accI = 0.f;
  const int nspan = NPTS / NSPLIT;         // 512
  const int nbase0 = s * nspan;

  for (int c0 = 0; c0 < nspan; c0 += CHUNK) {
    __syncthreads();
    const int nb = nbase0 + c0;
    const float ang = xp[(size_t)p * NPTS + nb + np] * sf;
    #pragma unroll
    for (int e = 0; e < 8; ++e) {
      const int j = q * 8 + e;             // j<16: E1[j]=e^{-2pi i j ang}; else E16[j-24*16..]
      const float th = (j < 16)
          ? (-kTWO_PI * (float)j * ang)
          : (-kTWO_PI * 16.0f * (float)(j - 24) * ang);   // k1 = (j-16)-8
      float sv, cv;
      sincosf(th, &sv, &cv);
      if (j < 16) sE1[np][j]       = make_float2(cv, sv);
      else        sE16[np][j - 16] = make_float2(cv, sv);
    }
    if (kk < CHUNK) sW[kk] = w[nb + kk];
    __syncthreads();

    #pragma unroll 4
    for (int n = 0; n < CHUNK; ++n) {
      const float2 a = sE16[n][k1i];
      const float2 b = sE1[n][k0];
      const float er = a.x * b.x - a.y * b.y;   // e^{-2pi i k ang_n}
      const float ei = a.x * b.y + a.y * b.x;
      const float wn = sW[n];
      accR = fmaf(wn, er, accR);
      accI = fmaf(wn, ei, accI);
    }
  }
  const int o = ((s * PSL + p) * KFTN + kk) * 2;
  apart[o] = accR;
  apart[o + 1] = accI;
}

// ---------------- K3: c[p,k] = kft(k) * sum_s apart --------------------------
__global__ void k_scalec(const float* __restrict__ apart, float* __restrict__ c,
                         const unsigned* __restrict__ maxbits,
                         const float* __restrict__ scale) {
  const int p = blockIdx.x;
  const int kk = threadIdx.x;
  const float sf = 0.3f / __uint_as_float(*maxbits);
  const float sigma = scale[0] * sf;
  const float s2 = sigma * sigma;
  const int k = kk - KFTN / 2;
  float kf = 0.f;
  if (k != 0) {
    const float ak = fabsf((float)k);
    const float lv = kLOGC + 31.f * logf(ak) + 16.f * logf(kTWO_PI * s2)
                     - kTWO_PI2 * s2 * (float)(k * k);
    kf = expf(lv);
  }
  float ar = 0.f, ai = 0.f;
  #pragma unroll
  for (int s = 0; s < NSPLIT; ++s) {
    const int o = ((s * PSL + p) * KFTN + kk) * 2;
    ar += apart[o];
    ai += apart[o + 1];
  }
  const int o = (p * KFTN + kk) * 2;
  c[o] = kf * ar;
  c[o + 1] = kf * ai;
}

// ---------------- K4: forward NUDFT via phase recurrence, c[p] in LDS ---------
__global__ void k_stageB(const float* __restrict__ yp, const float* __restrict__ c,
                         float* __restrict__ opart,
                         const unsigned* __restrict__ maxbits) {
  __shared__ float2 sc[KFTN];
  const int p = blockIdx.y;
  const int m = blockIdx.x * blockDim.x + threadIdx.x;
  sc[threadIdx.x] = ((const float2*)c)[p * KFTN + threadIdx.x];
  __syncthreads();

  const float sf = 0.3f / __uint_as_float(*maxbits);
  const float phi = kTWO_PI * yp[(size_t)p * MPTS + m] * sf;
  float br, bi, er, ei;
  sincosf(phi, &bi, &br);                  // base = e^{i phi}
  sincosf(-128.f * phi, &ei, &er);         // E = e^{i*(-128)*phi}

  float acc = 0.f;
  #pragma unroll 8
  for (int kk = 0; kk < KFTN; ++kk) {      // Re( c[k] * E_k ), E_{k+1} = E_k*base
    const float2 ck = sc[kk];
    acc = fmaf(ck.x, er, acc);
    acc = fmaf(-ck.y, ei, acc);
    const float nr = er * br - ei * bi;
    ei = er * bi + ei * br;
    er = nr;
  }
  opart[(size_t)p * MPTS + m] = acc;
}

// ---------------- K5: deterministic 1/P reduction -----------------------------
__global__ void k_reduce(const float* __restrict__ opart, float* __restrict__ out) {
  const int m = blockIdx.x * blockDim.x + threadIdx.x;
  float s = 0.f;
  #pragma unroll
  for (int p = 0; p < PSL; ++p) s += opart[(size_t)p * MPTS + m];
  out[m] = s * (1.0f / PSL);
}

// ---------------- launch ------------------------------------------------------
extern "C" void kernel_launch(void* const* d_in, const int* in_sizes, int n_in,
                              void* d_out, int out_size, void* d_ws, size_t ws_size,
                              hipStream_t stream) {
  (void)in_sizes; (void)n_in; (void)out_size; (void)ws_size;
  const float* x     = (const float*)d_in[0];
  const float* y     = (const float*)d_in[1];
  const float* w     = (const float*)d_in[2];
  const float* xis   = (const float*)d_in[3];
  const float* scale = (const float*)d_in[4];
  float* out = (float*)d_out;

  // workspace layout (floats)
  float* ws = (float*)d_ws;
  float*    xisn  = ws;                                   // 512
  float*    xp    = xisn + PSL * DIMD;                    // 65536
  float*    yp    = xp + (size_t)PSL * NPTS;              // 65536
  float*    apart = yp + (size_t)PSL * MPTS;              // 8*16*256*2 = 65536
  float*    c     = apart + (size_t)NSPLIT * PSL * KFTN * 2; // 8192
  float*    opart = c + (size_t)PSL * KFTN * 2;           // 65536
  unsigned* maxbits = (unsigned*)(opart + (size_t)PSL * MPTS); // 1

  k_init<<<1, 512, 0, stream>>>(xis, xisn, maxbits);
  k_proj<<<(NPTS / 16) + (MPTS / 16), 32, 0, stream>>>(xisn, x, y, xp, yp, maxbits);
  k_stageA<<<dim3(PSL, NSPLIT), KFTN, 0, stream>>>(xp, w, apart, maxbits);
  k_scalec<<<PSL, KFTN, 0, stream>>>(apart, c, maxbits, scale);
  k_stageB<<<dim3(MPTS / 256, PSL), 256, 0, stream>>>(yp, c, opart, maxbits);
  k_reduce<<<MPTS / 256, 256, 0, stream>>>(opart, out);
}